// NonlocalWeightedAverage_34084860461735
// MI455X (gfx1250) — compile-verified
//
#include <hip/hip_runtime.h>
#include <hip/hip_bf16.h>

// NonlocalWeightedAverage, fused for MI455X (gfx1250, wave32).
// corr = lf^T lf (K=576) via V_WMMA_F32_16X16X4_F32 with 4 independent
// accumulator tiles per wave. lf is staged in pair-packed layout
// lfP[b][k/2][n][2] so every B fragment is one global_load_b64 and every
// A fragment one ds_load_b64. 16-row score strip lives in 320KB WGP LDS;
// row-softmax(corr*10) + 3-channel weighted average are fused (float4 loads).

typedef __attribute__((ext_vector_type(2))) float v2f;
typedef __attribute__((ext_vector_type(8))) float v8f;

#define BATCH 4
#define CCH   64
#define HW    64
#define NPOS  4096            // 64*64
#define KDIM  576             // 64*9
#define KPAIR (KDIM / 2)      // 288
#define INV_ALPHA 10.0f

// ---------------------------------------------------------------------------
// Kernel 1: unfold3x3 -> pair-packed lfP[b][kp][n][2], k = 2*kp + k0,
//           k = c*9 + (dy+1)*3 + (dx+1)
// ---------------------------------------------------------------------------
__global__ __launch_bounds__(256)
void nlwa_unfold_kernel(const float* __restrict__ feat, float* __restrict__ lf) {
    int idx = blockIdx.x * 256 + threadIdx.x;      // < 4*288*4096 = 4718592
    int n  = idx & (NPOS - 1);
    int kp = (idx >> 12) % KPAIR;
    int b  = idx / (KPAIR * NPOS);
    float2 v;
    #pragma unroll
    for (int k0 = 0; k0 < 2; ++k0) {
        int k = (kp << 1) | k0;
        int c = k / 9;
        int p = k - c * 9;
        int y = (n >> 6) + (p / 3) - 1;
        int x = (n & 63) + (p % 3) - 1;
        float t = 0.0f;
        if ((unsigned)y < HW && (unsigned)x < HW)
            t = feat[(((size_t)b * CCH + c) << 12) + (y << 6) + x];
        if (k0) v.y = t; else v.x = t;
    }
    ((float2*)lf)[idx] = v;                        // b64 store, coalesced
}

// ---------------------------------------------------------------------------
// Kernel 2: per (batch, 16-row tile): GEMM -> LDS strip -> softmax -> out
// Dynamic LDS layout:
//   float scorr[16][NPOS]        262144 B   (score strip)
//   float2 lfA [KPAIR][16]        36864 B   (pair-packed A-panel)
// total 299008 B  (fits CDNA5's 320KB/WGP LDS)
// ---------------------------------------------------------------------------
extern __shared__ char nlwa_smem[];

__global__ __launch_bounds__(256)
void nlwa_corr_softmax_kernel(const float* __restrict__ lf,
                              const float* __restrict__ x_lab,
                              float* __restrict__ out) {
    float* scorr = (float*)nlwa_smem;              // 16 * NPOS floats
    v2f*   lfA2  = (v2f*)(scorr + 16 * NPOS);      // KPAIR * 16 pairs

    const int tile = blockIdx.x;                   // 0..1023
    const int b    = tile >> 8;
    const int n0   = (tile & 255) << 4;
    const v2f* lfB2 = (const v2f*)lf + (size_t)b * KPAIR * NPOS;

    const int tid = threadIdx.x;

    // ---- stage pair-packed A-panel into LDS (one b64 per element) ----
    for (int i = tid; i < KPAIR * 16; i += 256) {
        int kp = i >> 4;
        lfA2[i] = lfB2[(kp << 12) + n0 + (i & 15)];
    }
    __syncthreads();

    const int wave  = tid >> 5;      // 0..7
    const int lane  = tid & 31;
    const int Mn    = lane & 15;     // row (A) / col (B,C,D) index within tile
    const int khalf = lane >> 4;     // which K pair this half-wave holds

    // ---- GEMM: each wave owns 8 groups of 4 consecutive m-tiles ----
    // 4 independent accumulator chains; A fragment = 1 ds_load_b64,
    // each B fragment = 1 global_load_b64 (contiguous 128B per half-wave).
    for (int g = 0; g < 8; ++g) {
        const int m0 = ((g * 8 + wave) << 6);       // 4 tiles * 16 cols
        const v2f* bbase = lfB2 + m0 + Mn;          // + kb*NPOS per step
        v8f acc0 = {}, acc1 = {}, acc2 = {}, acc3 = {};
        #pragma unroll 2
        for (int kk = 0; kk < KDIM; kk += 4) {
            const int kb = (kk >> 1) + khalf;       // pair index (ka = 2*kb)
            // A 16x4 fragment: v0 <- K=2kb, v1 <- K=2kb+1 (one b64)
            const v2f av = lfA2[(kb << 4) + Mn];
            // B 4x16 fragments for 4 adjacent tiles (one b64 each)
            const v2f* bp = bbase + kb * NPOS;
            const v2f b0 = bp[0];
            const v2f b1 = bp[16];
            const v2f b2 = bp[32];
            const v2f b3 = bp[48];
            acc0 = __builtin_amdgcn_wmma_f32_16x16x4_f32(
                false, av, false, b0, (short)0, acc0, false, false);
            acc1 = __builtin_amdgcn_wmma_f32_16x16x4_f32(
                false, av, false, b1, (short)0, acc1, false, false);
            acc2 = __builtin_amdgcn_wmma_f32_16x16x4_f32(
                false, av, false, b2, (short)0, acc2, false, false);
            acc3 = __builtin_amdgcn_wmma_f32_16x16x4_f32(
                false, av, false, b3, (short)0, acc3, false, false);
        }
        // C/D layout: VGPR r holds row (r + 8*khalf), column = Mn
        const int rowbase = khalf << 3;
        #pragma unroll
        for (int r = 0; r < 8; ++r) {
            float* srow = scorr + ((rowbase + r) << 12) + m0 + Mn;
            srow[0]  = acc0[r];
            srow[16] = acc1[r];
            srow[32] = acc2[r];
            srow[48] = acc3[r];
        }
    }
    __syncthreads();

    // ---- fused softmax(corr*10) + weighted average, 2 rows per wave ----
    const float* xl = x_lab + (size_t)b * 3 * NPOS;
    const float4* x0 = (const float4*)xl;
    const float4* x1 = (const float4*)(xl + NPOS);
    const float4* x2 = (const float4*)(xl + 2 * NPOS);
    for (int rr = 0; rr < 2; ++rr) {
        const int row = (wave << 1) + rr;
        const float4* s4 = (const float4*)(scorr + (row << 12));

        float lmax = -3.4e38f;
        for (int q = lane; q < NPOS / 4; q += 32) {      // ds_load_b128
            float4 v = s4[q];
            lmax = fmaxf(lmax, fmaxf(fmaxf(v.x, v.y), fmaxf(v.z, v.w)));
        }
        #pragma unroll
        for (int off = 16; off > 0; off >>= 1)
            lmax = fmaxf(lmax, __shfl_xor(lmax, off, 32));

        float ss = 0.f, s0 = 0.f, s1 = 0.f, s2 = 0.f;
        for (int q = lane; q < NPOS / 4; q += 32) {
            float4 v = s4[q];
            float4 a = x0[q], bb = x1[q], cc = x2[q];    // global_load_b128
            float e0 = __expf((v.x - lmax) * INV_ALPHA);
            float e1 = __expf((v.y - lmax) * INV_ALPHA);
            float e2 = __expf((v.z - lmax) * INV_ALPHA);
            float e3 = __expf((v.w - lmax) * INV_ALPHA);
            ss += (e0 + e1) + (e2 + e3);
            s0 += e0 * a.x  + e1 * a.y  + e2 * a.z  + e3 * a.w;
            s1 += e0 * bb.x + e1 * bb.y + e2 * bb.z + e3 * bb.w;
            s2 += e0 * cc.x + e1 * cc.y + e2 * cc.z + e3 * cc.w;
        }
        #pragma unroll
        for (int off = 16; off > 0; off >>= 1) {
            ss += __shfl_xor(ss, off, 32);
            s0 += __shfl_xor(s0, off, 32);
            s1 += __shfl_xor(s1, off, 32);
            s2 += __shfl_xor(s2, off, 32);
        }
        if (lane == 0) {
            const float inv = 1.0f / ss;
            float* ob = out + (size_t)b * 3 * NPOS + n0 + row;
            ob[0]        = s0 * inv;
            ob[NPOS]     = s1 * inv;
            ob[2 * NPOS] = s2 * inv;
        }
    }
}

// ---------------------------------------------------------------------------
extern "C" void kernel_launch(void* const* d_in, const int* in_sizes, int n_in,
                              void* d_out, int out_size, void* d_ws, size_t ws_size,
                              hipStream_t stream) {
    const float* x_lab   = (const float*)d_in[0];   // [4,3,64,64]
    const float* feature = (const float*)d_in[1];   // [4,64,64,64]
    float* out = (float*)d_out;                     // [4,3,64,64]
    float* lf  = (float*)d_ws;                      // pair-packed, 36 MB

    (void)in_sizes; (void)n_in; (void)out_size; (void)ws_size;

    const int unfold_pairs = BATCH * KPAIR * NPOS;  // 4718592, multiple of 256
    nlwa_unfold_kernel<<<unfold_pairs / 256, 256, 0, stream>>>(feature, lf);

    const size_t lds_bytes = (size_t)(16 * NPOS + KDIM * 16) * sizeof(float);
    static int attr_set = 0;   // configuration only; identical every call
    if (!attr_set) {
        hipFuncSetAttribute((const void*)nlwa_corr_softmax_kernel,
                            hipFuncAttributeMaxDynamicSharedMemorySize,
                            (int)lds_bytes);
        attr_set = 1;
    }
    nlwa_corr_softmax_kernel<<<BATCH * (NPOS / 16), 256, lds_bytes, stream>>>(
        lf, x_lab, out);
}